// CTRNNPolicy_75496935129666
// MI455X (gfx1250) — compile-verified
//
#include <hip/hip_runtime.h>

typedef __attribute__((ext_vector_type(2))) float v2f;
typedef __attribute__((ext_vector_type(8))) float v8f;

#define B_    64
#define N_    1024
#define OBS_  64
#define ACT_  16
#define ITERS_ 10
#define DT_   0.1f

// One workgroup (32 waves, wave32) per batch element.
// Thread n owns row n of the CTRNN state (v, I, gain/bias/mask/tau in regs).
// Recurrent matvec W[b] @ y via V_WMMA_F32_16X16X4_F32:
//   wave w -> rows [32w, 32w+32), two 16-row accumulators c0/c1.
//   A tile: lane L supplies W[row0 + (L&15)][k + 2*(L>>4) + {0,1}]  (b64 load)
//   B tile: y[k + 2*(L>>4) + {0,1}] broadcast to all 16 columns (LDS b64)
//   => every D column equals the 16 row dot-products; lanes 0/16 write them out.
__launch_bounds__(1024, 1)
__global__ void ctrnn_policy_kernel(const float* __restrict__ obs,
                                    const float* __restrict__ v0,
                                    const float* __restrict__ tau,
                                    const float* __restrict__ gain,
                                    const float* __restrict__ bias,
                                    const float* __restrict__ W,
                                    const float* __restrict__ mask,
                                    const float* __restrict__ E,
                                    const float* __restrict__ Dm,
                                    float* __restrict__ out)
{
    __shared__ __align__(16) float obs_s[OBS_];
    __shared__ __align__(16) float y_s[N_];
    __shared__ __align__(16) float wy_s[N_];
    __shared__ __align__(16) float vf_s[N_];

    const int b    = blockIdx.x;
    const int n    = threadIdx.x;       // 0..1023 : row ownership
    const int lane = n & 31;
    const int wave = n >> 5;

    if (n < OBS_) obs_s[n] = obs[(size_t)b * OBS_ + n];

    // per-row parameters live in registers for the whole scan
    const size_t bn = (size_t)b * N_ + n;
    float vn  = v0[bn];
    float g   = gain[bn];
    float bi  = bias[bn];
    float mk  = mask[bn];
    float idt = DT_ / tau[bn];

    __syncthreads();

    // encode: I_n = E[b,n,:] . obs[b]   (kept in a register)
    float In = 0.0f;
    {
        const float* er = E + bn * OBS_;
        #pragma unroll
        for (int o = 0; o < OBS_; o += 4) {
            float4 e4 = *(const float4*)(er + o);
            float4 o4 = *(const float4*)(obs_s + o);
            In += e4.x * o4.x + e4.y * o4.y + e4.z * o4.z + e4.w * o4.w;
        }
    }

    // WMMA fragment mapping for this lane
    const int row0 = wave * 32;
    const int rsub = lane & 15;
    const int koff = (lane >> 4) * 2;
    const float* Wr0 = W + ((size_t)b * N_ + (size_t)(row0 + rsub)) * N_ + koff;
    const float* Wr1 = Wr0 + (size_t)16 * N_;

    for (int t = 0; t < ITERS_; ++t) {
        // masked tanh gate (row-local -> no barrier needed before write)
        y_s[n] = tanhf(g * (vn + bi)) * mk;
        __syncthreads();

        v8f c0 = {};
        v8f c1 = {};
        for (int kk = 0; kk < N_; kk += 128) {
            // stream-prefetch 1KB ahead on each W row (speculative, safe)
            __builtin_prefetch(Wr0 + kk + 256, 0, 1);
            __builtin_prefetch(Wr1 + kk + 256, 0, 1);
            #pragma unroll
            for (int k = kk; k < kk + 128; k += 4) {
                float2 yv = *(const float2*)(y_s + k + koff);
                v2f bb; bb.x = yv.x; bb.y = yv.y;

                float2 a0 = *(const float2*)(Wr0 + k);
                float2 a1 = *(const float2*)(Wr1 + k);
                v2f A0; A0.x = a0.x; A0.y = a0.y;
                v2f A1; A1.x = a1.x; A1.y = a1.y;

                c0 = __builtin_amdgcn_wmma_f32_16x16x4_f32(
                        false, A0, false, bb, (short)0, c0, false, false);
                c1 = __builtin_amdgcn_wmma_f32_16x16x4_f32(
                        false, A1, false, bb, (short)0, c1, false, false);
            }
        }

        // D layout: VGPR p holds M=p (lanes 0-15) / M=8+p (lanes 16-31);
        // all 16 columns identical -> lanes 0 and 16 hold the 16 results each.
        if ((lane & 15) == 0) {
            const int rb = row0 + (lane >> 4) * 8;
            float4* p0 = (float4*)(wy_s + rb);
            p0[0] = make_float4(c0[0], c0[1], c0[2], c0[3]);
            p0[1] = make_float4(c0[4], c0[5], c0[6], c0[7]);
            float4* p1 = (float4*)(wy_s + rb + 16);
            p1[0] = make_float4(c1[0], c1[1], c1[2], c1[3]);
            p1[1] = make_float4(c1[4], c1[5], c1[6], c1[7]);
        }
        __syncthreads();

        // leaky integration, row-local
        float dv = -vn + wy_s[n] + In;
        vn = (vn + idt * dv) * mk;
    }

    vf_s[n] = vn;
    __syncthreads();

    // decode: action[b,a] = D[b,a,:] . v   (16 tiny dot products)
    if (n < ACT_) {
        const float* dr = Dm + ((size_t)b * ACT_ + n) * N_;
        float acc = 0.0f;
        for (int k = 0; k < N_; k += 4) {
            float4 d4 = *(const float4*)(dr + k);
            float4 v4 = *(const float4*)(vf_s + k);
            acc += d4.x * v4.x + d4.y * v4.y + d4.z * v4.z + d4.w * v4.w;
        }
        out[(size_t)b * ACT_ + n] = acc;
    }
}

extern "C" void kernel_launch(void* const* d_in, const int* in_sizes, int n_in,
                              void* d_out, int out_size, void* d_ws, size_t ws_size,
                              hipStream_t stream) {
    const float* obs  = (const float*)d_in[0];
    const float* v0   = (const float*)d_in[1];
    const float* tau  = (const float*)d_in[2];
    const float* gain = (const float*)d_in[3];
    const float* bias = (const float*)d_in[4];
    const float* W    = (const float*)d_in[5];
    const float* mask = (const float*)d_in[6];
    const float* E    = (const float*)d_in[7];
    const float* Dm   = (const float*)d_in[8];
    float* out = (float*)d_out;

    hipLaunchKernelGGL(ctrnn_policy_kernel, dim3(B_), dim3(N_), 0, stream,
                       obs, v0, tau, gain, bias, W, mask, E, Dm, out);
}